// DDM_26491358282394
// MI455X (gfx1250) — compile-verified
//
#include <hip/hip_runtime.h>
#include <math.h>

// CDNA5 / gfx1250: fp32 WMMA 16x16x4 (wave32) + Tensor Data Mover staging.
typedef float v2f __attribute__((ext_vector_type(2)));
typedef float v8f __attribute__((ext_vector_type(8)));
typedef unsigned int u32x4 __attribute__((ext_vector_type(4)));
typedef int i32x4 __attribute__((ext_vector_type(4)));
typedef int i32x8 __attribute__((ext_vector_type(8)));

#if defined(__has_builtin)
#if __has_builtin(__builtin_amdgcn_tensor_load_to_lds)
#define HAVE_TDM 1
#endif
#endif

// Problem constants (from reference setup_inputs)
constexpr int B_TOT  = 2048;
constexpr int TF     = 120;
constexpr int DLAT   = 16;
constexpr int P      = 16;
constexpr int CHUNK  = 42;          // 1 + (M+1) + M, M=20
constexpr int OC     = CHUNK * P;   // 672 columns of h per timestep
constexpr int OUTC   = 43;          // 1 + 21 + 21 packed output per feature
constexpr int MROWS  = 32;          // batch rows per block
constexpr int HPAD   = 676;         // 672 + 4 pad -> conflict-free half-wave stores
constexpr int ZPAD   = 18;          // 16 + 2 pad -> 16 distinct banks, keeps 8B align
constexpr float INV_TAU = 10.0f;    // 1 / 0.1

__global__ __launch_bounds__(256)
void spline_wmma_kernel(const float* __restrict__ z,
                        const float* __restrict__ W,
                        const float* __restrict__ bias,
                        float* __restrict__ out) {
    __shared__ float h_lds[MROWS][HPAD];
    __shared__ float w_lds[DLAT * OC];   // 16 x 672 weight panel for this timestep
    __shared__ float z_lds[MROWS][ZPAD];

    const int t    = blockIdx.y;
    const int b0   = blockIdx.x * MROWS;
    const int tid  = threadIdx.x;
    const int lane = tid & 31;
    const int wave = tid >> 5;

    const float* Wt = W + (size_t)t * DLAT * OC;
    const float* bt = bias + (size_t)t * OC;

#ifdef HAVE_TDM
    // ---- TDM: DMA the 16x672 f32 weight panel (43KB) global -> LDS ----
    if (wave == 0) {
        const unsigned long long ga = (unsigned long long)(uintptr_t)Wt;
        const unsigned int ldsoff   = (unsigned int)(uintptr_t)&w_lds[0];
        u32x4 g0;
        g0.x = 1u;                                  // count=1, user descriptor
        g0.y = ldsoff;                              // lds_addr (bytes)
        g0.z = (unsigned int)ga;                    // global_addr[31:0]
        g0.w = (unsigned int)((ga >> 32) & 0x01FFFFFFu) | (2u << 30); // addr[56:32] | type=2
        i32x8 g1;
        g1[0] = 0x00020000;                         // wg_mask=0, data_size=2 (4 bytes)
        g1[1] = (int)(672u << 16);                  // tensor_dim0 = 672 (lo16 at [31:16])
        g1[2] = (int)(16u  << 16);                  // tensor_dim0 hi=0 | tensor_dim1 = 16
        g1[3] = (int)(672u << 16);                  // tensor_dim1 hi=0 | tile_dim0 = 672
        g1[4] = 16;                                 // tile_dim1 = 16, tile_dim2 = 0
        g1[5] = 672;                                // tensor_dim0_stride lo32 = 672
        g1[6] = 0;                                  // stride hi | tensor_dim1_stride lo
        g1[7] = 0;
        i32x4 zz = {0, 0, 0, 0};
#if __clang_major__ >= 23
        i32x8 z8 = {0, 0, 0, 0, 0, 0, 0, 0};
        __builtin_amdgcn_tensor_load_to_lds(g0, g1, zz, zz, z8, 0);
#else
        __builtin_amdgcn_tensor_load_to_lds(g0, g1, zz, zz, 0);
#endif
    }
#else
    // Fallback: cooperative vector copy of the weight panel.
    {
        const float4* src = (const float4*)Wt;
        float4* dst = (float4*)&w_lds[0];
        for (int i = tid; i < DLAT * OC / 4; i += 256) dst[i] = src[i];
    }
#endif

    // Stage the 32x16 z tile into LDS (overlaps the TDM transfer).
    for (int i = tid; i < MROWS * DLAT; i += 256) {
        int r = i >> 4, d = i & 15;
        z_lds[r][d] = z[((size_t)(b0 + r) * TF + t) * DLAT + d];
    }

#ifdef HAVE_TDM
    if (wave == 0) __builtin_amdgcn_s_wait_tensorcnt(0);
#endif
    __syncthreads();

    // ---- GEMM phase: h[32 x 672] = z_tile[32 x 16] * W[t][16 x 672] + b[t] ----
    // 84 output tiles (2 M-tiles x 42 N-tiles) round-robined over 8 waves.
    const int lrow = lane & 15;        // M (A) / N (B) index within tile
    const int koff = (lane >> 4) * 2;  // lanes 0-15 -> K={0,1}; lanes 16-31 -> K={2,3}

    for (int tile = wave; tile < 2 * 42; tile += 8) {
        const int mt  = tile / 42;
        const int nt  = tile % 42;
        const int col = nt * 16 + lrow;

        v8f c = {0.f, 0.f, 0.f, 0.f, 0.f, 0.f, 0.f, 0.f};
#pragma unroll
        for (int kb = 0; kb < DLAT; kb += 4) {
            v2f a, bf;
            a.x  = z_lds[mt * 16 + lrow][kb + koff];
            a.y  = z_lds[mt * 16 + lrow][kb + koff + 1];
            bf.x = w_lds[(kb + koff)     * OC + col];
            bf.y = w_lds[(kb + koff + 1) * OC + col];
            // D = A(16x4,f32) * B(4x16,f32) + C ; 4 chained steps cover K=16.
            c = __builtin_amdgcn_wmma_f32_16x16x4_f32(
                    false, a, false, bf, (short)0, c, false, false);
        }

        const float bv    = bt[col];                   // bias is per-column
        const int   rbase = mt * 16 + (lane >> 4) * 8; // VGPR j: M=j (lo), M=8+j (hi)
#pragma unroll
        for (int j = 0; j < 8; ++j)
            h_lds[rbase + j][nt * 16 + lrow] = c[j] + bv;
    }
    __syncthreads();

    // ---- Nonlinearity + pack phase: per (row, feature) spline params ----
    for (int item = tid; item < MROWS * P; item += 256) {
        const int r = item >> 4;
        const int p = item & 15;
        const float* hr = &h_lds[r][p * CHUNK];
        float* o = out + ((size_t)(b0 + r) * TF + t) * (P * OUTC) + (size_t)p * OUTC;

        // gamma
        o[0] = hr[0];

        // beta = softplus(h[1 : 22])  (21 values), stable: max(x,0)+log(1+e^-|x|)
#pragma unroll
        for (int j = 0; j < 21; ++j) {
            float x = hr[1 + j];
            o[1 + j] = fmaxf(x, 0.f) + __logf(1.f + __expf(-fabsf(x)));
        }

        // s = softmax(h[22 : 42] / tau); delta = [0, cumsum(s)]  (21 values)
        float v[20];
        float m = -3.4e38f;
#pragma unroll
        for (int j = 0; j < 20; ++j) { v[j] = hr[22 + j]; m = fmaxf(m, v[j]); }
        float s = 0.f;
#pragma unroll
        for (int j = 0; j < 20; ++j) { v[j] = __expf((v[j] - m) * INV_TAU); s += v[j]; }
        const float inv = 1.0f / s;
        float acc = 0.f;
        o[22] = 0.f;
#pragma unroll
        for (int j = 0; j < 20; ++j) { acc += v[j]; o[23 + j] = acc * inv; }
    }
}

extern "C" void kernel_launch(void* const* d_in, const int* in_sizes, int n_in,
                              void* d_out, int out_size, void* d_ws, size_t ws_size,
                              hipStream_t stream) {
    const float* z    = (const float*)d_in[0];  // [2048, 120, 16]
    const float* W    = (const float*)d_in[1];  // [120, 16, 672]
    const float* bias = (const float*)d_in[2];  // [120, 672]
    float* out = (float*)d_out;                 // [2048, 120, 16, 43]

    dim3 grid(B_TOT / MROWS, TF);
    dim3 block(256);
    spline_wmma_kernel<<<grid, block, 0, stream>>>(z, W, bias, out);
}